// NoisyTopItemsPerExpertRouter_3719441678674
// MI455X (gfx1250) — compile-verified
//
#include <hip/hip_runtime.h>

typedef __attribute__((ext_vector_type(2))) float v2f;
typedef __attribute__((ext_vector_type(8))) float v8f;

#define NUM_B 4
#define SEQ   2048
#define DIM   1024
#define NEXP  64
#define CAP   64
#define M_ROWS (NUM_B * SEQ)              /* 8192 rows */
#define NOISE_SCALE (1.0f / 64.0f)        /* NOISE_STD / NUM_EXPERTS */

// ---------------------------------------------------------------------------
// Kernel 0: transpose W [D,E] -> Wt [E,D] so WMMA B-fragments are contiguous
// ---------------------------------------------------------------------------
__global__ void transpose_w(const float* __restrict__ W, float* __restrict__ Wt) {
  int i = blockIdx.x * blockDim.x + threadIdx.x;   // over D*E
  if (i < DIM * NEXP) {
    int k = i / NEXP;
    int n = i % NEXP;
    Wt[n * DIM + k] = W[i];
  }
}

// ---------------------------------------------------------------------------
// Kernel 1: fused gate GEMM (f32 WMMA) + noise + softmax over E=64
//   one wave -> 16 rows x 64 experts (four 16x16 accumulators)
//   V_WMMA_F32_16X16X4_F32: A 16x4 (lane: m=lane&15, k pair = 2*(lane>>4)),
//   B 4x16 mirrored (n=lane&15), D 16x16 (vgpr r: M=r / r+8, N=lane&15)
// ---------------------------------------------------------------------------
__global__ __launch_bounds__(128)
void gemm_softmax(const float* __restrict__ X, const float* __restrict__ Wt,
                  const float* __restrict__ noise, float* __restrict__ gates) {
  const int wave = threadIdx.x >> 5;
  const int lane = threadIdx.x & 31;
  const int half = lane >> 4;
  const int l    = lane & 15;
  const int row0 = (blockIdx.x * 4 + wave) * 16;

  v8f acc0 = {}, acc1 = {}, acc2 = {}, acc3 = {};

  const float* arow = X  + (size_t)(row0 + l) * DIM + 2 * half;
  const float* b0   = Wt + (size_t)( 0 + l) * DIM + 2 * half;
  const float* b1   = Wt + (size_t)(16 + l) * DIM + 2 * half;
  const float* b2   = Wt + (size_t)(32 + l) * DIM + 2 * half;
  const float* b3   = Wt + (size_t)(48 + l) * DIM + 2 * half;

  for (int k0 = 0; k0 < DIM; k0 += 4) {
    v2f a  = *(const v2f*)(arow + k0);
    v2f w0 = *(const v2f*)(b0 + k0);
    v2f w1 = *(const v2f*)(b1 + k0);
    v2f w2 = *(const v2f*)(b2 + k0);
    v2f w3 = *(const v2f*)(b3 + k0);
    acc0 = __builtin_amdgcn_wmma_f32_16x16x4_f32(false, a, false, w0, (short)0, acc0, false, false);
    acc1 = __builtin_amdgcn_wmma_f32_16x16x4_f32(false, a, false, w1, (short)0, acc1, false, false);
    acc2 = __builtin_amdgcn_wmma_f32_16x16x4_f32(false, a, false, w2, (short)0, acc2, false, false);
    acc3 = __builtin_amdgcn_wmma_f32_16x16x4_f32(false, a, false, w3, (short)0, acc3, false, false);
  }

  // In-register softmax per row. Row (r, r+8) pairs reduce independently via
  // xor-butterfly masks 1,2,4,8 (never cross the 16-lane halves).
  #pragma unroll
  for (int r = 0; r < 8; ++r) {
    const int row = row0 + r + 8 * half;
    const float* nrow = noise + (size_t)row * NEXP + l;
    float v0 = acc0[r] + nrow[0]  * NOISE_SCALE;
    float v1 = acc1[r] + nrow[16] * NOISE_SCALE;
    float v2 = acc2[r] + nrow[32] * NOISE_SCALE;
    float v3 = acc3[r] + nrow[48] * NOISE_SCALE;

    float mx = fmaxf(fmaxf(v0, v1), fmaxf(v2, v3));
    mx = fmaxf(mx, __shfl_xor(mx, 1, 32));
    mx = fmaxf(mx, __shfl_xor(mx, 2, 32));
    mx = fmaxf(mx, __shfl_xor(mx, 4, 32));
    mx = fmaxf(mx, __shfl_xor(mx, 8, 32));

    v0 = __expf(v0 - mx); v1 = __expf(v1 - mx);
    v2 = __expf(v2 - mx); v3 = __expf(v3 - mx);
    float sum = v0 + v1 + v2 + v3;
    sum += __shfl_xor(sum, 1, 32);
    sum += __shfl_xor(sum, 2, 32);
    sum += __shfl_xor(sum, 4, 32);
    sum += __shfl_xor(sum, 8, 32);
    float inv = 1.0f / sum;

    float* grow = gates + (size_t)row * NEXP + l;
    grow[0]  = v0 * inv;
    grow[16] = v1 * inv;
    grow[32] = v2 * inv;
    grow[48] = v3 * inv;
  }
}

// ---------------------------------------------------------------------------
// Kernel 2: zero the 268 MB output with b128 stores (the real bandwidth cost)
// ---------------------------------------------------------------------------
__global__ void zero_out(float4* __restrict__ out, int n4) {
  int i = blockIdx.x * blockDim.x + threadIdx.x;
  int stride = gridDim.x * blockDim.x;
  float4 z = make_float4(0.f, 0.f, 0.f, 0.f);
  for (; i < n4; i += stride) out[i] = z;
}

// ---------------------------------------------------------------------------
// Kernel 3: per-(b,e) top-64 of 2048 gates, scatter mask/combine non-zeros.
// Iterative argmax with tie-break on lower token index (jax top_k semantics).
// ---------------------------------------------------------------------------
__global__ __launch_bounds__(256)
void topk_scatter(const float* __restrict__ gates, float* __restrict__ dispatch,
                  float* __restrict__ combine) {
  __shared__ float vals[SEQ];
  __shared__ float red_v[256];
  __shared__ int   red_i[256];
  const int tid = threadIdx.x;
  const int b = blockIdx.x >> 6;
  const int e = blockIdx.x & 63;

  for (int s = tid; s < SEQ; s += 256)
    vals[s] = gates[(size_t)(b * SEQ + s) * NEXP + e];
  __syncthreads();

  for (int c = 0; c < CAP; ++c) {
    float bv = -3.0e38f; int bi = SEQ;
    for (int s = tid; s < SEQ; s += 256) {
      float v = vals[s];
      if (v > bv) { bv = v; bi = s; }   // strided scan keeps lowest s per thread
    }
    red_v[tid] = bv; red_i[tid] = bi;
    __syncthreads();
    for (int st = 128; st > 0; st >>= 1) {
      if (tid < st) {
        float ov = red_v[tid + st]; int oi = red_i[tid + st];
        float cv = red_v[tid];      int ci = red_i[tid];
        if (ov > cv || (ov == cv && oi < ci)) { red_v[tid] = ov; red_i[tid] = oi; }
      }
      __syncthreads();
    }
    if (tid == 0) {
      int s = red_i[0];
      size_t base = (((size_t)(b * SEQ + s) * NEXP) + e) * CAP + c;
      dispatch[base] = 1.0f;
      combine[base]  = red_v[0];
      vals[s] = -3.0e38f;               // exclude for next round
    }
    __syncthreads();
  }
}

// ---------------------------------------------------------------------------
extern "C" void kernel_launch(void* const* d_in, const int* in_sizes, int n_in,
                              void* d_out, int out_size, void* d_ws, size_t ws_size,
                              hipStream_t stream) {
  (void)in_sizes; (void)n_in; (void)ws_size;
  const float* X     = (const float*)d_in[0];   // [B,S,D]
  const float* W     = (const float*)d_in[1];   // [D,E]
  const float* noise = (const float*)d_in[2];   // [B,S,E]

  float* out      = (float*)d_out;
  float* dispatch = out;                                        // [B,S,E,C]
  float* combine  = out + (size_t)M_ROWS * NEXP * CAP;          // [B,S,E,C]

  char*  ws    = (char*)d_ws;
  float* Wt    = (float*)ws;                                    // 256 KB
  float* gates = (float*)(ws + (size_t)DIM * NEXP * sizeof(float)); // 2 MB

  transpose_w<<<(DIM * NEXP + 255) / 256, 256, 0, stream>>>(W, Wt);
  gemm_softmax<<<M_ROWS / 64, 128, 0, stream>>>(X, Wt, noise, gates);
  zero_out<<<8192, 256, 0, stream>>>((float4*)d_out, out_size / 4);
  topk_scatter<<<NUM_B * NEXP, 256, 0, stream>>>(gates, dispatch, combine);
}